// OuterProductMean_33560874451053
// MI455X (gfx1250) — compile-verified
//
#include <hip/hip_runtime.h>

// ---------------------------------------------------------------------------
// OuterProductMean for MI455X (gfx1250, wave32).
//   s = seq @ W1 + b1                  [1024, 32]
//   t[i,c,p] = sum_d s[i,d] W2[c,d,p]  [1024, 32, 64]   (8 MB, L2-resident)
//   out[i,j,p] = sum_c t[i,c,p] s[j,c] + b2[p] + pair[i,j,p]
// Main contraction uses V_WMMA_F32_16X16X4_F32: A = s-tile [16j x 4c],
// B = t-tile [4c x 16p], 8 WMMAs accumulate K=32. pair/out streamed NT.
// ---------------------------------------------------------------------------

typedef __attribute__((ext_vector_type(2))) float v2f;
typedef __attribute__((ext_vector_type(8))) float v8f;

#define IN_DIM   256
#define DIM_MSA  32
#define PAIR_DIM 64
#define SEQ_L    1024

// ---- Kernel 1: s[i,d] = sum_k seq[i,k] * W1[k,d] + b1[d] ------------------
// One thread per (i,d); a wave covers all 32 d of one i, so seq[i,k] is a
// wave-uniform (scalar) load and W1[k,*] is a 128B coalesced load.
__global__ void __launch_bounds__(256)
opm_proj_kernel(const float* __restrict__ seq, const float* __restrict__ W1,
                const float* __restrict__ b1, float* __restrict__ s) {
    int tid = blockIdx.x * blockDim.x + threadIdx.x;   // 1024*32 threads
    int d = tid & (DIM_MSA - 1);
    int i = tid >> 5;
    const float* row = seq + (size_t)i * IN_DIM;
    float acc = 0.0f;
#pragma unroll 8
    for (int k = 0; k < IN_DIM; ++k)
        acc = fmaf(row[k], W1[k * DIM_MSA + d], acc);
    s[tid] = acc + b1[d];
}

// ---- Kernel 2: t[i,c,p] = sum_d s[i,d] * W2[(c*32+d)*64 + p] --------------
// One thread per (i,c,p); lanes vary over p -> coalesced W2 reads, s[i,d]
// is wave-uniform. W2 (256 KB) stays hot in cache.
__global__ void __launch_bounds__(256)
opm_tmat_kernel(const float* __restrict__ s, const float* __restrict__ W2,
                float* __restrict__ t) {
    int tid = blockIdx.x * blockDim.x + threadIdx.x;   // 1024*32*64 threads
    int p = tid & (PAIR_DIM - 1);
    int c = (tid >> 6) & (DIM_MSA - 1);
    int i = tid >> 11;
    const float* srow = s + (size_t)i * DIM_MSA;
    const float* wcol = W2 + ((size_t)c * DIM_MSA) * PAIR_DIM + p;
    float acc = 0.0f;
#pragma unroll 8
    for (int d = 0; d < DIM_MSA; ++d)
        acc = fmaf(srow[d], wcol[(size_t)d * PAIR_DIM], acc);
    t[tid] = acc;
}

// ---- Kernel 3: main streaming WMMA contraction ----------------------------
// Block = 256 threads = 8 waves, handles one i and a 32-row j-chunk:
//   wave w -> j-tile (w>>2), p-tile (w&3); each wave one 16x16 (j,p) tile.
// A-fragment layout (16x4 f32, ISA 7.12.2): v0 = K0 (lanes 0-15) / K2
// (lanes 16-31), v1 = K1 / K3 -> a single float2 load per K-step.
// B-fragment mirrors it from t[i] ([c][p], p across lanes 0-15).
// C/D layout: VGPR r -> row (r, r+8) split across lane halves, N=p per lane.
__global__ void __launch_bounds__(256)
opm_outer_kernel(const float* __restrict__ s, const float* __restrict__ t,
                 const float* __restrict__ b2, const float* __restrict__ pair,
                 float* __restrict__ out) {
    int i      = blockIdx.x >> 5;        // 1024 values of i
    int jchunk = blockIdx.x & 31;        // 32 chunks of 32 j-rows
    int wave = threadIdx.x >> 5;
    int lane = threadIdx.x & 31;
    int half = lane >> 4;                // 0: lanes 0-15, 1: lanes 16-31
    int l    = lane & 15;
    int jbase = jchunk * 32 + (wave >> 2) * 16;
    int pbase = (wave & 3) * 16;

    // A source: s rows jbase..jbase+15, contraction dim c = 0..31
    const float* srow = s + (size_t)(jbase + l) * DIM_MSA + 2 * half;
    // B source: t[i], rows c, cols pbase..pbase+15
    const float* tcol = t + ((size_t)i * DIM_MSA + 2 * half) * PAIR_DIM + pbase + l;

    v8f acc = {};
#pragma unroll
    for (int k = 0; k < 8; ++k) {        // c = 4k .. 4k+3 per WMMA
        v2f a = *(const v2f*)(srow + 4 * k);           // c = 4k+2h, 4k+2h+1
        v2f b;
        b.x = tcol[(size_t)(4 * k) * PAIR_DIM];        // c = 4k+2h
        b.y = tcol[(size_t)(4 * k + 1) * PAIR_DIM];    // c = 4k+2h+1
        acc = __builtin_amdgcn_wmma_f32_16x16x4_f32(
            /*neg_a=*/false, a, /*neg_b=*/false, b,
            /*c_mod=*/(short)0, acc, /*reuse_a=*/false, /*reuse_b=*/false);
    }

    float bias = b2[pbase + l];
    // Epilogue: out[i, jbase + r + 8*half, pbase + l] for r = 0..7.
    size_t base = ((size_t)i * SEQ_L + jbase + 8 * half) * PAIR_DIM + pbase + l;
#pragma unroll
    for (int r = 0; r < 8; ++r) {
        size_t g = base + (size_t)r * PAIR_DIM;
        float pv = __builtin_nontemporal_load(pair + g);   // stream 256 MB in
        __builtin_nontemporal_store(acc[r] + bias + pv, out + g);  // 256 MB out
    }
}

extern "C" void kernel_launch(void* const* d_in, const int* in_sizes, int n_in,
                              void* d_out, int out_size, void* d_ws, size_t ws_size,
                              hipStream_t stream) {
    const float* seq  = (const float*)d_in[0];  // [1, 1024, 256]
    const float* pair = (const float*)d_in[1];  // [1, 1024, 1024, 64]
    const float* W1   = (const float*)d_in[2];  // [256, 32]
    const float* b1   = (const float*)d_in[3];  // [32]
    const float* W2   = (const float*)d_in[4];  // [1024, 64]
    const float* b2   = (const float*)d_in[5];  // [64]
    float* out = (float*)d_out;                 // [1, 1024, 1024, 64]

    float* s = (float*)d_ws;                    // 1024*32      = 128 KB
    float* t = s + (size_t)SEQ_L * DIM_MSA;     // 1024*32*64   = 8 MB

    opm_proj_kernel<<<(SEQ_L * DIM_MSA) / 256, 256, 0, stream>>>(seq, W1, b1, s);
    opm_tmat_kernel<<<(SEQ_L * DIM_MSA * PAIR_DIM) / 256, 256, 0, stream>>>(s, W2, t);
    opm_outer_kernel<<<SEQ_L * 32, 256, 0, stream>>>(s, t, b2, pair, out);
}